// PoolWrapper_84636625535188
// MI455X (gfx1250) — compile-verified
//
#include <hip/hip_runtime.h>
#include <stdint.h>

// Segment mean-pool, MI455X (gfx1250, wave32).
// Memory-bound: 512MB stream of x -> async global->LDS double-buffered pipeline,
// NT temporal hint, register accumulation over sorted segments, rare f32 atomics.

#define DIMS 64
#define TILE_ROWS 16                         // rows per LDS tile
#define TILES_PER_CHUNK 32
#define CHUNK (TILE_ROWS * TILES_PER_CHUNK)  // 512 rows per wave
#define COPIES 8                             // 8 x (32 lanes x 16B) = 4096B tile

__device__ __forceinline__ unsigned lds_off(const void* p) {
  // generic LDS pointer = {shared-aperture hi32, as3-offset lo32}
  return (unsigned)(uintptr_t)p;
}

// Fast path: one (lds, global) base pair; IOFFSET applies to BOTH sides of the
// async copy (CDNA5 ISA), so 8 issues differ only in offset imm.
template <int I>
__device__ __forceinline__ void issue8(unsigned lbase, unsigned long long gbase) {
  asm volatile("global_load_async_to_lds_b128 %0, %1, off offset:%2 th:TH_LOAD_NT"
               :: "v"(lbase), "v"(gbase), "i"(I * 512) : "memory");
  if constexpr (I + 1 < COPIES) issue8<I + 1>(lbase, gbase);
}

__global__ void __launch_bounds__(32)
segsum_kernel(const float* __restrict__ x, const int* __restrict__ batch,
              float* __restrict__ out, long long N) {
  __shared__ __align__(16) float lbuf[2][TILE_ROWS * DIMS];
  const int lane = threadIdx.x;
  const long long chunk_base = (long long)blockIdx.x * CHUNK;
  if (chunk_base >= N) return;
  const long long chunk_end = (chunk_base + CHUNK < N) ? (chunk_base + CHUNK) : N;
  const int nt = (int)((chunk_end - chunk_base + TILE_ROWS - 1) / TILE_ROWS);

  const unsigned long long xbase = (unsigned long long)(uintptr_t)x;
  const long long max_goff = N * DIMS - 4;  // clamp (in float elements), 16B-chunk safe

  auto issue_tile = [&](int t, int b) {
    const long long tile_base = chunk_base + (long long)t * TILE_ROWS;
    const unsigned lbase = lds_off(&lbuf[b][0]) + lane * 16;
    const long long g0 = tile_base * DIMS + lane * 4;
    if (tile_base + TILE_ROWS <= N) {
      // full tile: no clamping, offset-immediate form (uniform branch)
      issue8<0>(lbase, xbase + (unsigned long long)g0 * 4ull);
    } else {
      // final partial tile only: per-16B clamp; clamped rows are never consumed
#pragma unroll
      for (int i = 0; i < COPIES; ++i) {
        long long goff = g0 + (long long)i * 128;
        if (goff > max_goff) goff = max_goff;
        unsigned long long gaddr = xbase + (unsigned long long)goff * 4ull;
        unsigned laddr = lbase + i * 512;
        asm volatile("global_load_async_to_lds_b128 %0, %1, off th:TH_LOAD_NT"
                     :: "v"(laddr), "v"(gaddr) : "memory");
      }
    }
  };

  issue_tile(0, 0);

  int cur_seg = -1;
  float ax = 0.f, ay = 0.f;
  const long long col = (long long)lane * 2;

  for (int t = 0; t < nt; ++t) {
    if (t + 1 < nt) {
      // buffer (t+1)&1 was last read at iteration t-1; make sure those ds reads retired
      asm volatile("s_wait_dscnt 0x0" ::: "memory");
      issue_tile(t + 1, (t + 1) & 1);
      asm volatile("s_wait_asynccnt 0x8" ::: "memory");  // tile t's 8 copies done (in-order)
    } else {
      asm volatile("s_wait_asynccnt 0x0" ::: "memory");
    }
    const int b = t & 1;
    const long long tile_base = chunk_base + (long long)t * TILE_ROWS;

    long long brow = tile_base + (lane & 15);
    if (brow >= N) brow = N - 1;
    int seg_l = batch[brow];  // lanes 0..15 hold this tile's segment ids

    long long rem = chunk_end - tile_base;
    int rows = rem < TILE_ROWS ? (int)rem : TILE_ROWS;
    const float2* rowp = (const float2*)&lbuf[b][lane * 2];
#pragma unroll 4
    for (int r = 0; r < rows; ++r) {
      int seg = __builtin_amdgcn_readlane(seg_l, r);  // SGPR broadcast, scalar compare
      if (seg != cur_seg) {                           // scalar branch: segment boundary
        if (cur_seg >= 0) {
          atomicAdd(&out[(long long)cur_seg * DIMS + col + 0], ax);
          atomicAdd(&out[(long long)cur_seg * DIMS + col + 1], ay);
        }
        cur_seg = seg; ax = 0.f; ay = 0.f;
      }
      float2 v = rowp[r * (DIMS / 2)];  // ds_load_b64, bank-conflict free
      ax += v.x; ay += v.y;
    }
  }
  if (cur_seg >= 0) {
    atomicAdd(&out[(long long)cur_seg * DIMS + col + 0], ax);
    atomicAdd(&out[(long long)cur_seg * DIMS + col + 1], ay);
  }
}

__global__ void zero_kernel(float* __restrict__ p, long long n) {
  long long i = (long long)blockIdx.x * blockDim.x + threadIdx.x;
  if (i < n) p[i] = 0.f;
}

// counts via binary search on sorted batch (no extra pass over x, no workspace)
__global__ void __launch_bounds__(DIMS)
finalize_kernel(const int* __restrict__ batch, long long N, float* __restrict__ out) {
  const int g = blockIdx.x;
  const int d = threadIdx.x;
  __shared__ float inv;
  if (d == 0) {
    long long lo = 0, hi = N;
    while (lo < hi) { long long m = (lo + hi) >> 1; if (batch[m] < g) lo = m + 1; else hi = m; }
    long long s = lo;
    hi = N;
    while (lo < hi) { long long m = (lo + hi) >> 1; if (batch[m] < g + 1) lo = m + 1; else hi = m; }
    long long c = lo - s;
    inv = 1.0f / (float)(c > 0 ? c : 1);
  }
  __syncthreads();
  out[(long long)g * DIMS + d] *= inv;
}

extern "C" void kernel_launch(void* const* d_in, const int* in_sizes, int n_in,
                              void* d_out, int out_size, void* d_ws, size_t ws_size,
                              hipStream_t stream) {
  const float* x = (const float*)d_in[0];
  const int* batch = (const int*)d_in[1];
  const long long N = (long long)in_sizes[1];
  float* out = (float*)d_out;
  const int G = out_size / DIMS;

  const long long total_out = (long long)out_size;
  zero_kernel<<<(int)((total_out + 255) / 256), 256, 0, stream>>>(out, total_out);

  const long long nchunks = (N + CHUNK - 1) / CHUNK;
  segsum_kernel<<<(int)nchunks, 32, 0, stream>>>(x, batch, out, N);

  finalize_kernel<<<G, DIMS, 0, stream>>>(batch, N, out);
}